// KernelConvolution_48816598286978
// MI455X (gfx1250) — compile-verified
//
#include <hip/hip_runtime.h>
#include <math.h>

// ---------------------------------------------------------------------------
// FNO spectral conv on gfx1250: 5 GEMM stages using V_WMMA_F32_16X16X4_F32,
// with LDS staging (padded rows, conflict-free fragment reads) for all
// row-gathered operands. B=32, C=64, X=T=128, M1=M2=32, NS = B*C = 2048.
// ---------------------------------------------------------------------------

typedef __attribute__((ext_vector_type(2))) float v2f;
typedef __attribute__((ext_vector_type(8))) float v8f;

__device__ __forceinline__ v8f wmma4(v2f a, v2f b, v8f c) {
  return __builtin_amdgcn_wmma_f32_16x16x4_f32(false, a, false, b, (short)0, c,
                                               false, false);
}

// --- twiddle tables --------------------------------------------------------
// Wt[t][m] = e^{-2pi i * t*(m-16)/128}        (forward DFT over t, 128x32)
// Fx[m][x] = e^{-2pi i * x*(m-16)/128}        (forward DFT over x, 32x128)
// Gx[x][m] = e^{+2pi i * x*(m-16)/128}/128    (inverse over x, 128x32)
// Gt[m][t] = e^{+2pi i * t*(m-16)/128}/128    (inverse over t, 32x128)
__global__ void k_twiddle(float* __restrict__ wt_re, float* __restrict__ wt_im,
                          float* __restrict__ fx_re, float* __restrict__ fx_im,
                          float* __restrict__ gx_re, float* __restrict__ gx_im,
                          float* __restrict__ gt_re, float* __restrict__ gt_im) {
  int idx = blockIdx.x * blockDim.x + threadIdx.x;  // 0..4095
  int i0 = idx >> 5;   // 0..127 : x or t
  int i1 = idx & 31;   // 0..31  : mode m
  float theta = 3.14159265358979323846f * (float)(i0 * (i1 - 16)) / 64.0f;
  float s, c;
  __sincosf(theta, &s, &c);
  const float inv = 1.0f / 128.0f;
  wt_re[i0 * 32 + i1] = c;        wt_im[i0 * 32 + i1] = -s;
  fx_re[i1 * 128 + i0] = c;       fx_im[i1 * 128 + i0] = -s;
  gx_re[i0 * 32 + i1] = c * inv;  gx_im[i0 * 32 + i1] = s * inv;
  gt_re[i1 * 128 + i0] = c * inv; gt_im[i1 * 128 + i0] = s * inv;
}

// --- Stage 1: S1[s][x][m2] = z[s][x][:] @ Wt -------------------------------
// one block (256 thr = 8 waves) per slice; z slice staged in LDS (pad 129);
// wave wv owns row-strip tx=wv and computes both m2 column-tiles, reusing A.
__global__ void k_stage1(const float* __restrict__ z,
                         const float* __restrict__ wt_re,
                         const float* __restrict__ wt_im,
                         float* __restrict__ s1_re, float* __restrict__ s1_im) {
  __shared__ float Az[128 * 129];
  int slice = blockIdx.x;
  const float4* src = (const float4*)(z + slice * 16384);
  for (int i = 0; i < 16; ++i) {
    int idx4 = threadIdx.x + i * 256;          // 0..4095 (32 float4 per row)
    float4 v = src[idx4];
    int row = idx4 >> 5, c4 = (idx4 & 31) << 2;
    float* d = &Az[row * 129 + c4];
    d[0] = v.x; d[1] = v.y; d[2] = v.z; d[3] = v.w;
  }
  __syncthreads();
  int wv = threadIdx.x >> 5, lane = threadIdx.x & 31;
  int lo = lane & 15, hi = lane >> 4, kk = hi * 2;
  int rA = wv * 16 + lo;
  v8f ar0 = {}, ai0 = {}, ar1 = {}, ai1 = {};
  for (int k0 = 0; k0 < 128; k0 += 4) {
    int ka = k0 + kk;
    v2f a;   a.x = Az[rA * 129 + ka];            a.y = Az[rA * 129 + ka + 1];
    v2f br0; br0.x = wt_re[ka * 32 + lo];        br0.y = wt_re[(ka + 1) * 32 + lo];
    v2f bi0; bi0.x = wt_im[ka * 32 + lo];        bi0.y = wt_im[(ka + 1) * 32 + lo];
    v2f br1; br1.x = wt_re[ka * 32 + 16 + lo];   br1.y = wt_re[(ka + 1) * 32 + 16 + lo];
    v2f bi1; bi1.x = wt_im[ka * 32 + 16 + lo];   bi1.y = wt_im[(ka + 1) * 32 + 16 + lo];
    ar0 = wmma4(a, br0, ar0);  ai0 = wmma4(a, bi0, ai0);
    ar1 = wmma4(a, br1, ar1);  ai1 = wmma4(a, bi1, ai1);
  }
  int base0 = slice * 4096 + lo;
  for (int v = 0; v < 8; ++v) {
    int r = wv * 16 + v + 8 * hi;
    s1_re[base0 + r * 32] = ar0[v];       s1_im[base0 + r * 32] = ai0[v];
    s1_re[base0 + 16 + r * 32] = ar1[v];  s1_im[base0 + 16 + r * 32] = ai1[v];
  }
}

// --- Stage 2: S2[s][m1][m2] = Fx @ S1[s] (complex) -------------------------
// one block (128 thr = 4 waves) per slice, one tile per wave; Fx staged.
__global__ void k_stage2(const float* __restrict__ s1_re,
                         const float* __restrict__ s1_im,
                         const float* __restrict__ fx_re,
                         const float* __restrict__ fx_im,
                         float* __restrict__ s2_re, float* __restrict__ s2_im) {
  __shared__ float FxR[32 * 129];
  __shared__ float FxI[32 * 129];
  int slice = blockIdx.x;
  for (int i = 0; i < 8; ++i) {
    int idx4 = threadIdx.x + i * 128;          // 0..1023
    float4 vr = ((const float4*)fx_re)[idx4];
    float4 vi = ((const float4*)fx_im)[idx4];
    int row = idx4 >> 5, c4 = (idx4 & 31) << 2;
    float* dr = &FxR[row * 129 + c4];
    dr[0] = vr.x; dr[1] = vr.y; dr[2] = vr.z; dr[3] = vr.w;
    float* di = &FxI[row * 129 + c4];
    di[0] = vi.x; di[1] = vi.y; di[2] = vi.z; di[3] = vi.w;
  }
  __syncthreads();
  int wv = threadIdx.x >> 5, lane = threadIdx.x & 31;
  int lo = lane & 15, hi = lane >> 4, kk = hi * 2;
  int tm1 = wv >> 1, tm2 = wv & 1;
  int rA = tm1 * 16 + lo, cB = tm2 * 16 + lo;
  const float* Br = s1_re + slice * 4096;
  const float* Bi = s1_im + slice * 4096;
  v8f RR = {}, II = {}, RI = {}, IR = {};
  for (int k0 = 0; k0 < 128; k0 += 4) {
    int ka = k0 + kk;
    v2f are; are.x = FxR[rA * 129 + ka];  are.y = FxR[rA * 129 + ka + 1];
    v2f aim; aim.x = FxI[rA * 129 + ka];  aim.y = FxI[rA * 129 + ka + 1];
    v2f bre; bre.x = Br[ka * 32 + cB];    bre.y = Br[(ka + 1) * 32 + cB];
    v2f bim; bim.x = Bi[ka * 32 + cB];    bim.y = Bi[(ka + 1) * 32 + cB];
    RR = wmma4(are, bre, RR);  II = wmma4(aim, bim, II);
    RI = wmma4(are, bim, RI);  IR = wmma4(aim, bre, IR);
  }
  int base = slice * 1024 + tm2 * 16 + lo;
  for (int v = 0; v < 8; ++v) {
    int r = tm1 * 16 + v + 8 * hi;
    s2_re[base + r * 32] = RR[v] - II[v];
    s2_im[base + r * 32] = RI[v] + IR[v];
  }
}

// --- Stage 3: per-mode channel mix  S3[b][j] = sum_i S2[b][i]*w[i][j] ------
// one block (256 thr = 8 waves) per mode; strided panels gathered to LDS
// once, then all fragment reads are conflict-free LDS reads.
__global__ void k_stage3(const float* __restrict__ s2_re,
                         const float* __restrict__ s2_im,
                         const float* __restrict__ wr,
                         const float* __restrict__ wi,
                         float* __restrict__ s3_re, float* __restrict__ s3_im) {
  __shared__ float AsR[32 * 65];
  __shared__ float AsI[32 * 65];
  __shared__ float BsR[64 * 64];
  __shared__ float BsI[64 * 64];
  int mode = blockIdx.x;
  for (int i = 0; i < 8; ++i) {            // A panel: 2048 gathered elems
    int e = threadIdx.x + i * 256;         // e = b*64 + ii
    int b = e >> 6, ii = e & 63;
    AsR[b * 65 + ii] = s2_re[e * 1024 + mode];
    AsI[b * 65 + ii] = s2_im[e * 1024 + mode];
  }
  for (int i = 0; i < 16; ++i) {           // B panel: 4096 gathered elems
    int e = threadIdx.x + i * 256;         // e = ii*64 + j
    BsR[e] = wr[e * 1024 + mode];
    BsI[e] = wi[e * 1024 + mode];
  }
  __syncthreads();
  int wv = threadIdx.x >> 5, lane = threadIdx.x & 31;
  int lo = lane & 15, hi = lane >> 4, kk = hi * 2;
  int tb = wv >> 2, tj = wv & 3;
  int rowb = tb * 16 + lo, colj = tj * 16 + lo;
  v8f RR = {}, II = {}, RI = {}, IR = {};
  for (int k0 = 0; k0 < 64; k0 += 4) {
    int i0 = k0 + kk;
    v2f are; are.x = AsR[rowb * 65 + i0];     are.y = AsR[rowb * 65 + i0 + 1];
    v2f aim; aim.x = AsI[rowb * 65 + i0];     aim.y = AsI[rowb * 65 + i0 + 1];
    v2f bre; bre.x = BsR[i0 * 64 + colj];     bre.y = BsR[(i0 + 1) * 64 + colj];
    v2f bim; bim.x = BsI[i0 * 64 + colj];     bim.y = BsI[(i0 + 1) * 64 + colj];
    RR = wmma4(are, bre, RR);  II = wmma4(aim, bim, II);
    RI = wmma4(are, bim, RI);  IR = wmma4(aim, bre, IR);
  }
  for (int v = 0; v < 8; ++v) {
    int idx = ((tb * 16 + v + 8 * hi) * 64 + colj) * 1024 + mode;
    s3_re[idx] = RR[v] - II[v];
    s3_im[idx] = RI[v] + IR[v];
  }
}

// --- Stage 4: S4[s][x][m2] = Gx @ S3[s] (complex, K=32) --------------------
// one block (256 thr = 8 waves) per slice; Gx staged (pad 33);
// wave wv owns row-strip tx=wv and both m2 tiles.
__global__ void k_stage4(const float* __restrict__ s3_re,
                         const float* __restrict__ s3_im,
                         const float* __restrict__ gx_re,
                         const float* __restrict__ gx_im,
                         float* __restrict__ s4_re, float* __restrict__ s4_im) {
  __shared__ float GxR[128 * 33];
  __shared__ float GxI[128 * 33];
  for (int i = 0; i < 4; ++i) {
    int idx4 = threadIdx.x + i * 256;          // 0..1023 (8 float4 per row)
    float4 vr = ((const float4*)gx_re)[idx4];
    float4 vi = ((const float4*)gx_im)[idx4];
    int row = idx4 >> 3, c4 = (idx4 & 7) << 2;
    float* dr = &GxR[row * 33 + c4];
    dr[0] = vr.x; dr[1] = vr.y; dr[2] = vr.z; dr[3] = vr.w;
    float* di = &GxI[row * 33 + c4];
    di[0] = vi.x; di[1] = vi.y; di[2] = vi.z; di[3] = vi.w;
  }
  __syncthreads();
  int slice = blockIdx.x;
  int wv = threadIdx.x >> 5, lane = threadIdx.x & 31;
  int lo = lane & 15, hi = lane >> 4, kk = hi * 2;
  int rA = wv * 16 + lo;
  const float* Br = s3_re + slice * 1024;
  const float* Bi = s3_im + slice * 1024;
  for (int t = 0; t < 2; ++t) {
    int cB = t * 16 + lo;
    v8f RR = {}, II = {}, RI = {}, IR = {};
    for (int k0 = 0; k0 < 32; k0 += 4) {
      int ka = k0 + kk;
      v2f are; are.x = GxR[rA * 33 + ka];  are.y = GxR[rA * 33 + ka + 1];
      v2f aim; aim.x = GxI[rA * 33 + ka];  aim.y = GxI[rA * 33 + ka + 1];
      v2f bre; bre.x = Br[ka * 32 + cB];   bre.y = Br[(ka + 1) * 32 + cB];
      v2f bim; bim.x = Bi[ka * 32 + cB];   bim.y = Bi[(ka + 1) * 32 + cB];
      RR = wmma4(are, bre, RR);  II = wmma4(aim, bim, II);
      RI = wmma4(are, bim, RI);  IR = wmma4(aim, bre, IR);
    }
    int base = slice * 4096 + t * 16 + lo;
    for (int v = 0; v < 8; ++v) {
      int r = wv * 16 + v + 8 * hi;
      s4_re[base + r * 32] = RR[v] - II[v];
      s4_im[base + r * 32] = RI[v] + IR[v];
    }
  }
}

// --- Stage 5: out[s][x][t] = Re(S4[s] @ Gt) --------------------------------
// one block (256 thr = 8 waves) per slice; S4 slice staged (pad 33);
// wave wv owns row-strip tx=wv, loops the 8 t-column tiles.
__global__ void k_stage5(const float* __restrict__ s4_re,
                         const float* __restrict__ s4_im,
                         const float* __restrict__ gt_re,
                         const float* __restrict__ gt_im,
                         float* __restrict__ out) {
  __shared__ float AsR[128 * 33];
  __shared__ float AsI[128 * 33];
  int slice = blockIdx.x;
  const float4* srcR = (const float4*)(s4_re + slice * 4096);
  const float4* srcI = (const float4*)(s4_im + slice * 4096);
  for (int i = 0; i < 4; ++i) {
    int idx4 = threadIdx.x + i * 256;          // 0..1023
    float4 vr = srcR[idx4];
    float4 vi = srcI[idx4];
    int row = idx4 >> 3, c4 = (idx4 & 7) << 2;
    float* dr = &AsR[row * 33 + c4];
    dr[0] = vr.x; dr[1] = vr.y; dr[2] = vr.z; dr[3] = vr.w;
    float* di = &AsI[row * 33 + c4];
    di[0] = vi.x; di[1] = vi.y; di[2] = vi.z; di[3] = vi.w;
  }
  __syncthreads();
  int wv = threadIdx.x >> 5, lane = threadIdx.x & 31;
  int lo = lane & 15, hi = lane >> 4, kk = hi * 2;
  int rA = wv * 16 + lo;
  for (int tt = 0; tt < 8; ++tt) {
    int cB = tt * 16 + lo;
    v8f RR = {}, II = {};
    for (int k0 = 0; k0 < 32; k0 += 4) {
      int ka = k0 + kk;
      v2f are; are.x = AsR[rA * 33 + ka];       are.y = AsR[rA * 33 + ka + 1];
      v2f aim; aim.x = AsI[rA * 33 + ka];       aim.y = AsI[rA * 33 + ka + 1];
      v2f bre; bre.x = gt_re[ka * 128 + cB];    bre.y = gt_re[(ka + 1) * 128 + cB];
      v2f bim; bim.x = gt_im[ka * 128 + cB];    bim.y = gt_im[(ka + 1) * 128 + cB];
      RR = wmma4(are, bre, RR);
      II = wmma4(aim, bim, II);
    }
    int base = slice * 16384 + cB;
    for (int v = 0; v < 8; ++v) {
      int r = wv * 16 + v + 8 * hi;
      out[base + r * 128] = RR[v] - II[v];
    }
  }
}

extern "C" void kernel_launch(void* const* d_in, const int* in_sizes, int n_in,
                              void* d_out, int out_size, void* d_ws, size_t ws_size,
                              hipStream_t stream) {
  const float* z  = (const float*)d_in[0];   // (32,64,128,128)
  const float* wr = (const float*)d_in[1];   // (64,64,32,32)
  const float* wi = (const float*)d_in[2];   // (64,64,32,32)
  float* out = (float*)d_out;
  float* ws = (float*)d_ws;

  // workspace layout (floats)
  float* wt_re = ws;            // 128*32
  float* wt_im = ws + 4096;
  float* fx_re = ws + 8192;     // 32*128
  float* fx_im = ws + 12288;
  float* gx_re = ws + 16384;    // 128*32
  float* gx_im = ws + 20480;
  float* gt_re = ws + 24576;    // 32*128
  float* gt_im = ws + 28672;
  float* s1_re = ws + 32768;               // 2048*128*32 = 8388608
  float* s1_im = s1_re + 8388608;
  float* s2_re = s1_im + 8388608;          // 2048*32*32 = 2097152
  float* s2_im = s2_re + 2097152;
  float* s3_re = s2_im + 2097152;
  float* s3_im = s3_re + 2097152;
  float* s4_re = s1_re;  // S1 dead after stage 2 — reuse
  float* s4_im = s1_im;

  k_twiddle<<<16, 256, 0, stream>>>(wt_re, wt_im, fx_re, fx_im,
                                    gx_re, gx_im, gt_re, gt_im);
  k_stage1<<<2048, 256, 0, stream>>>(z, wt_re, wt_im, s1_re, s1_im);
  k_stage2<<<2048, 128, 0, stream>>>(s1_re, s1_im, fx_re, fx_im, s2_re, s2_im);
  k_stage3<<<1024, 256, 0, stream>>>(s2_re, s2_im, wr, wi, s3_re, s3_im);
  k_stage4<<<2048, 256, 0, stream>>>(s3_re, s3_im, gx_re, gx_im, s4_re, s4_im);
  k_stage5<<<2048, 256, 0, stream>>>(s4_re, s4_im, gt_re, gt_im, out);
}